// FuzzyLogic_57501022158955
// MI455X (gfx1250) — compile-verified
//
#include <hip/hip_runtime.h>

typedef float v2f __attribute__((ext_vector_type(2)));
typedef float v4f __attribute__((ext_vector_type(4)));
typedef float v8f __attribute__((ext_vector_type(8)));

#define BATCH            16384
#define MAZE_DIM         3844      // 31*31*4 floats per row
#define CELLS            961       // 31*31 cells of 4 floats
#define WAVES_PER_BLOCK  8
#define BLOCK_THREADS    256       // 8 wave32
#define GRID1            (BATCH / WAVES_PER_BLOCK)   // 2048 blocks
#define CLAMP_LOG        90.0f     // -log(exp(-90))

// ---------------------------------------------------------------------------
// Per-cell class table, built at compile time.
//   0: dead cell (only (30,28))
//   1: OR pair     term = uv/(u+v-2uv), u=1-t0, v=1-t1
//   2: tile1 single, weight 1:  term = (1-t1)/t1
//   3: tile1 single, weight 2 (cell (30,0) is duplicated in IDX1)
//   4: tile0 single ((1,1),(29,29))
//   5: tile2 single ((0,1))
//   6: tile3 single ((30,29))
// ---------------------------------------------------------------------------
struct CellTbl { unsigned char v[CELLS]; };

constexpr CellTbl make_tbl() {
    CellTbl t{};
    for (int cell = 0; cell < CELLS; ++cell) {
        const int r = cell / 31;
        const int c = cell % 31;
        const bool pair = (r >= 1 && r <= 29 && c >= 1 && c <= 29) &&
                          !(r == 1 && c == 1) && !(r == 29 && c == 29);
        int w1 = 0;
        if (c == 0)  ++w1;
        if (c == 30) ++w1;
        if (r == 0  && c >= 2 && c <= 29) ++w1;
        if (r == 30 && c <= 27)           ++w1;
        unsigned char code = 0;
        if (pair)                                   code = 1;
        else if (w1 == 1)                           code = 2;
        else if (w1 == 2)                           code = 3;
        else if ((r == 1 && c == 1) || (r == 29 && c == 29)) code = 4;
        else if (r == 0 && c == 1)                  code = 5;
        else if (r == 30 && c == 29)                code = 6;
        t.v[cell] = code;
    }
    return t;
}

__constant__ CellTbl g_tbl = make_tbl();

// ---------------------------------------------------------------------------
// Kernel 1: one wave per batch row. Stream the full row with NT b128 loads,
// accumulate S = f(c1)+f(c2) per lane, reduce 32 lanes with one WMMA
// (A = ones 16x4 f32), emit per-block partial of min(log1p(S), 90).
// ---------------------------------------------------------------------------
__global__ __launch_bounds__(BLOCK_THREADS)
void maze_rows_kernel(const float* __restrict__ maze, float* __restrict__ partial) {
    const int lane = threadIdx.x & 31;
    const int wave = threadIdx.x >> 5;
    const int row  = blockIdx.x * WAVES_PER_BLOCK + wave;
    const v4f* rp  = (const v4f*)(maze + (size_t)row * MAZE_DIM);

    float s = 0.0f;
    for (int cell = lane; cell < CELLS; cell += 32) {
        const v4f t = __builtin_nontemporal_load(rp + cell);
        const unsigned code = g_tbl.v[cell];

        const float u = 1.0f - t.x;
        const float v = 1.0f - t.y;
        const float p = u * v;

        float tk = t.y;                      // codes 2,3
        if (code == 4u) tk = t.x;
        if (code == 5u) tk = t.z;
        if (code == 6u) tk = t.w;
        const float w = (code == 3u) ? 2.0f : 1.0f;

        const bool isPair   = (code == 1u);
        const bool isSingle = (code >= 2u);
        const float N = isPair ? p                      : (isSingle ? w * (1.0f - tk) : 0.0f);
        const float D = isPair ? fmaf(-2.0f, p, u + v)  : (isSingle ? tk              : 1.0f);
        s += N * __builtin_amdgcn_rcpf(D);
    }

    // 32-lane sum via v_wmma_f32_16x16x4_f32: A = ones(16x4), B holds {s,0}
    // per lane. D[m,n] = sum_k B[k,n]; summing one row of D = sum of all
    // 64 B slots = sum over lanes of s (layout independent).
    v2f a; a.x = 1.0f; a.y = 1.0f;
    v2f b; b.x = s;    b.y = 0.0f;
    v8f acc = {};
    acc = __builtin_amdgcn_wmma_f32_16x16x4_f32(false, a, false, b, (short)0,
                                                acc, false, false);
    float rsum = acc[0];
    rsum += __shfl_xor(rsum, 8, 32);
    rsum += __shfl_xor(rsum, 4, 32);
    rsum += __shfl_xor(rsum, 2, 32);
    rsum += __shfl_xor(rsum, 1, 32);
    // -log(max(1/(1+S), e^-90)) == min(log1p(S), 90)

    __shared__ float wterm[WAVES_PER_BLOCK];
    if (lane == 0) wterm[wave] = fminf(log1pf(rsum), CLAMP_LOG);
    __syncthreads();
    if (threadIdx.x == 0) {
        float bs = 0.0f;
#pragma unroll
        for (int i = 0; i < WAVES_PER_BLOCK; ++i) bs += wterm[i];
        partial[blockIdx.x] = bs;
    }
}

// ---------------------------------------------------------------------------
// Kernel 2: single wave. Deterministic WMMA-chain reduction of the 2048
// block partials (64 values per v_wmma_f32_16x16x4_f32, accumulated in C),
// then 4 shuffles, scale by 1/BATCH, store the scalar mean.
// ---------------------------------------------------------------------------
__global__ __launch_bounds__(32)
void reduce_kernel(const float* __restrict__ partial, float* __restrict__ out) {
    const int lane = threadIdx.x;
    v2f ones; ones.x = 1.0f; ones.y = 1.0f;
    v8f acc = {};
    const v2f* p2 = (const v2f*)partial;
    for (int tile = 0; tile < GRID1 / 64; ++tile) {
        v2f b = p2[tile * 32 + lane];
        acc = __builtin_amdgcn_wmma_f32_16x16x4_f32(false, ones, false, b,
                                                    (short)0, acc, false, false);
    }
    float t = acc[0];
    t += __shfl_xor(t, 8, 32);
    t += __shfl_xor(t, 4, 32);
    t += __shfl_xor(t, 2, 32);
    t += __shfl_xor(t, 1, 32);
    if (lane == 0) out[0] = t * (1.0f / (float)BATCH);
}

// ---------------------------------------------------------------------------
extern "C" void kernel_launch(void* const* d_in, const int* in_sizes, int n_in,
                              void* d_out, int out_size, void* d_ws, size_t ws_size,
                              hipStream_t stream) {
    (void)in_sizes; (void)n_in; (void)out_size; (void)ws_size;
    const float* maze    = (const float*)d_in[0];
    float*       partial = (float*)d_ws;          // 2048 floats of scratch

    maze_rows_kernel<<<GRID1, BLOCK_THREADS, 0, stream>>>(maze, partial);
    reduce_kernel<<<1, 32, 0, stream>>>(partial, (float*)d_out);
}